// CrossEntropyLossWithProb_20083267076110
// MI455X (gfx1250) — compile-verified
//
#include <hip/hip_runtime.h>
#include <hip/hip_bf16.h>

// CDNA5 (gfx1250) NLL-loss-from-probabilities.
//
// reference:  -mean( log( max( probs[b, labels[b]], 1e-8 ) ) )
//
// Only the 8192 gathered elements are needed (≈32 KB), never the full 1 GB
// matrix -> the kernel is latency/launch bound, so we optimize for:
//   * non-temporal gather loads (single-use data, don't pollute the 192 MB L2)
//   * wave32 reduction on the matrix pipe: V_WMMA_F32_16X16X4_F32 with a
//     ones B-matrix sums the 32 per-lane partials in exact f32
//   * fixed-order (deterministic) two-stage reduction, no float atomics

typedef __attribute__((ext_vector_type(2))) float v2f;
typedef __attribute__((ext_vector_type(8))) float v8f;

#define NLL_THREADS 256
#define NLL_WAVES   (NLL_THREADS / 32)
#define NLL_MAXBLK  256
#define NLL_CLIP    1e-8f

__global__ __launch_bounds__(NLL_THREADS)
void nll_partial_kernel(const float* __restrict__ probs,
                        const int*   __restrict__ labels,
                        float*       __restrict__ partials,
                        int B, int V)
{
    __shared__ float s_wave[NLL_WAVES];

    const int tid    = blockIdx.x * NLL_THREADS + threadIdx.x;
    const int stride = gridDim.x * NLL_THREADS;

    // Per-thread partial: sum of log(max(p, eps)) over assigned rows.
    float acc = 0.0f;
    for (int i = tid; i < B; i += stride) {
        const int   lab = labels[i];
        const float p   = __builtin_nontemporal_load(
                              probs + (size_t)i * (size_t)V + (size_t)lab);
        acc += logf(fmaxf(p, NLL_CLIP));
    }

    // ---- Wave32 reduction via the matrix pipe ---------------------------
    // A (16x4 f32, 2 VGPRs/lane): lane m (m<16)   -> A[m][0] = acc
    //                             lane m+16       -> A[m][2] = acc
    //                             (K=1 and K=3 slots zero)
    // B (4x16) = all ones  =>  D[m][n] = acc(lane m) + acc(lane m+16)
    v2f a; a.x = acc;  a.y = 0.0f;
    v2f b; b.x = 1.0f; b.y = 1.0f;
    v8f c = {};
    // 8 args: (neg_a, A, neg_b, B, c_mod, C, reuse_a, reuse_b)
    c = __builtin_amdgcn_wmma_f32_16x16x4_f32(
            /*neg_a=*/false, a, /*neg_b=*/false, b,
            /*c_mod=*/(short)0, c, /*reuse_a=*/false, /*reuse_b=*/false);

    // C/D layout: VGPR r, lanes 0-15 -> D[r][lane];  lanes 16-31 -> D[8+r][..]
    // Summing c[0..7] in lanes 0-15 covers source lanes {0..7, 16..23};
    // in lanes 16-31 it covers {8..15, 24..31}.  xor-16 combine finishes it.
    float half = ((c[0] + c[1]) + (c[2] + c[3]))
               + ((c[4] + c[5]) + (c[6] + c[7]));
    float wave_total = half + __shfl_xor(half, 16, 32);

    // ---- Cross-wave, fixed-order block reduction ------------------------
    const int lane = threadIdx.x & 31;
    const int wave = threadIdx.x >> 5;
    if (lane == 0) s_wave[wave] = wave_total;
    __syncthreads();

    if (threadIdx.x == 0) {
        float bsum = 0.0f;
        #pragma unroll
        for (int w = 0; w < NLL_WAVES; ++w) bsum += s_wave[w];
        partials[blockIdx.x] = bsum;
    }
}

__global__ void nll_final_kernel(const float* __restrict__ partials,
                                 float*       __restrict__ out,
                                 int nparts, float inv_b)
{
    if (blockIdx.x == 0 && threadIdx.x == 0) {
        float s = 0.0f;
        for (int i = 0; i < nparts; ++i) s += partials[i];  // fixed order
        out[0] = -s * inv_b;
    }
}

extern "C" void kernel_launch(void* const* d_in, const int* in_sizes, int n_in,
                              void* d_out, int out_size, void* d_ws, size_t ws_size,
                              hipStream_t stream)
{
    const float* probs  = (const float*)d_in[0];
    const int*   labels = (const int*)d_in[1];   // per harness: integer -> const int*
    float*       out    = (float*)d_out;
    float*       parts  = (float*)d_ws;

    const int B = in_sizes[1];                 // 8192
    const int V = (int)(in_sizes[0] / (size_t)B ? in_sizes[0] / B : 0); // 32000
    const int Vs = (B > 0) ? (in_sizes[0] / B) : 0;

    int nblocks = (B + NLL_THREADS - 1) / NLL_THREADS;   // 32 for B=8192
    if (nblocks > NLL_MAXBLK) nblocks = NLL_MAXBLK;
    if (nblocks < 1) nblocks = 1;

    nll_partial_kernel<<<nblocks, NLL_THREADS, 0, stream>>>(probs, labels, parts, B, Vs);
    nll_final_kernel<<<1, 32, 0, stream>>>(parts, out, nblocks, 1.0f / (float)B);
}